// Sampler_18803366822376
// MI455X (gfx1250) — compile-verified
//
#include <hip/hip_runtime.h>
#include <cstdint>

// Problem constants (fixed by the reference).
#define VOCAB   128256
#define V4      (VOCAB / 4)          // 32064 float4 per row
#define SPLIT   16                   // blocks per row
#define CHUNK4  (V4 / SPLIT)         // 2004 float4 per block (exact)
#define THREADS 256                  // 8 wave32
#define NWAVE   (THREADS / 32)
#define NIT     ((CHUNK4 + THREADS - 1) / THREADS)   // 8 pipeline steps

// Map float bits to a monotonically ordered unsigned key.
__device__ __forceinline__ unsigned fkey(float x) {
    unsigned b = __float_as_uint(x);
    unsigned m = (unsigned)((int)b >> 31);     // all-ones if negative
    return b ^ (m | 0x80000000u);
}

// CDNA5 async global->LDS copy, 16 bytes per lane. Tracked by ASYNCcnt.
__device__ __forceinline__ void async_copy16(unsigned lds_off, const void* gaddr) {
    asm volatile("global_load_async_to_lds_b128 %0, %1, off"
                 :: "v"(lds_off), "v"((unsigned long long)(uintptr_t)gaddr)
                 : "memory");
}

__global__ __launch_bounds__(THREADS)
void sampler_main(const float* __restrict__ logits,
                  const float* __restrict__ temps,
                  const float* __restrict__ u,
                  unsigned long long* __restrict__ ws) {
    // Wave-private double-buffered tiles: [wave][buf][arr(logits/u)][lane] = 16 KB
    __shared__ float4 tile[NWAVE][2][2][32];
    __shared__ unsigned long long red[2][NWAVE];

    const int row  = blockIdx.y;
    const int tid  = threadIdx.x;
    const int w    = tid >> 5;
    const int lane = tid & 31;
    const int base4 = blockIdx.x * CHUNK4;
    const int end4  = base4 + CHUNK4;

    const float4* gl = (const float4*)logits + (size_t)row * V4;
    const float4* gu = (const float4*)u      + (size_t)row * V4;

    const float t    = temps[row];
    const float invT = 1.0f / fmaxf(t, 1e-6f);
    const float NLN2 = -0.6931471805599453f;   // fold -log into log2 * -ln2

    const unsigned lds_l[2] = { (unsigned)(uintptr_t)&tile[w][0][0][lane],
                                (unsigned)(uintptr_t)&tile[w][1][0][lane] };
    const unsigned lds_u[2] = { (unsigned)(uintptr_t)&tile[w][0][1][lane],
                                (unsigned)(uintptr_t)&tile[w][1][1][lane] };

    auto issue = [&](int it) {
        int f = base4 + it * THREADS + tid;
        f = f < end4 ? f : end4 - 1;           // clamp: stay in-bounds, masked later
        const int b = it & 1;
        async_copy16(lds_l[b], gl + f);
        async_copy16(lds_u[b], gu + f);
    };

    float bg = -__builtin_inff(); unsigned ig = 0;   // greedy arm
    float bs = -__builtin_inff(); unsigned is = 0;   // sampled arm

    issue(0);
    for (int it = 0; it < NIT; ++it) {
        if (it + 1 < NIT) {
            issue(it + 1);                                    // prefetch next tile
            asm volatile("s_wait_asynccnt 0x2" ::: "memory"); // drain tile `it`
        } else {
            asm volatile("s_wait_asynccnt 0x0" ::: "memory");
        }
        const int f = base4 + it * THREADS + tid;
        const int b = it & 1;
        const float4 lv = tile[w][b][0][lane];
        const float4 uv = tile[w][b][1][lane];
        if (f < end4) {
            const float lvs[4] = { lv.x, lv.y, lv.z, lv.w };
            const float uvs[4] = { uv.x, uv.y, uv.z, uv.w };
            #pragma unroll
            for (int j = 0; j < 4; ++j) {
                const unsigned idx = (unsigned)(f * 4 + j);
                // nan_to_num: NaN -> 0, +/-inf -> +/-FLT_MAX
                float v = lvs[j];
                v = __builtin_isnan(v) ? 0.0f
                                       : fminf(fmaxf(v, -3.402823466e38f), 3.402823466e38f);
                if (v > bg) { bg = v; ig = idx; }   // strict > keeps first index in-lane
                // gumbel = -log(-log(clip(u)))
                float uc  = fminf(fmaxf(uvs[j], 1e-10f), 1.0f - 1e-7f);
                float nl  = __log2f(uc) * NLN2;                  // -ln(uc) > 0
                float gmb = __log2f(nl) * NLN2;                  // -ln(-ln(uc))
                float sv  = __builtin_fmaf(v, invT, gmb);        // scaled + gumbel
                if (sv > bs) { bs = sv; is = idx; }
            }
        }
    }

    // Packed keys: (ordered float bits << 32) | ~idx  => u64 max == (value, then lowest idx)
    unsigned long long pg = ((unsigned long long)fkey(bg) << 32) | (unsigned)(~ig);
    unsigned long long ps = ((unsigned long long)fkey(bs) << 32) | (unsigned)(~is);

    #pragma unroll
    for (int off = 16; off > 0; off >>= 1) {      // wave32 butterfly reduce
        unsigned long long og = __shfl_xor(pg, off, 32);
        unsigned long long os = __shfl_xor(ps, off, 32);
        pg = og > pg ? og : pg;
        ps = os > ps ? os : ps;
    }
    if (lane == 0) { red[0][w] = pg; red[1][w] = ps; }
    __syncthreads();
    if (tid == 0) {
        unsigned long long g = red[0][0], s = red[1][0];
        #pragma unroll
        for (int i = 1; i < NWAVE; ++i) {
            g = red[0][i] > g ? red[0][i] : g;
            s = red[1][i] > s ? red[1][i] : s;
        }
        atomicMax(&ws[2 * row + 0], g);   // global_atomic_max_u64
        atomicMax(&ws[2 * row + 1], s);
    }
}

__global__ void sampler_final(const float* __restrict__ temps,
                              const unsigned long long* __restrict__ ws,
                              int* __restrict__ out, int B) {
    int b = blockIdx.x * blockDim.x + threadIdx.x;
    if (b < B) {
        unsigned long long p = (temps[b] <= 1e-6f) ? ws[2 * b + 0] : ws[2 * b + 1];
        out[b] = (int)(~(unsigned)(p & 0xFFFFFFFFull));
    }
}

extern "C" void kernel_launch(void* const* d_in, const int* in_sizes, int n_in,
                              void* d_out, int out_size, void* d_ws, size_t ws_size,
                              hipStream_t stream) {
    const float* logits = (const float*)d_in[0];
    const float* temps  = (const float*)d_in[1];
    const float* u      = (const float*)d_in[2];
    const int B = in_sizes[1];                       // temperatures length

    unsigned long long* ws = (unsigned long long*)d_ws;
    hipMemsetAsync(ws, 0, sizeof(unsigned long long) * 2 * (size_t)B, stream);

    dim3 grid(SPLIT, B, 1);
    sampler_main<<<grid, THREADS, 0, stream>>>(logits, temps, u, ws);
    sampler_final<<<(B + 127) / 128, 128, 0, stream>>>(temps, ws, (int*)d_out, B);
}